// RTDETR_postprocess_30554397344458
// MI455X (gfx1250) — compile-verified
//
#include <hip/hip_runtime.h>
#include <hip/hip_bf16.h>

// ---------------------------------------------------------------------------
// RT-DETR post-process (scale, normalize, sort, NMS, mask) for gfx1250.
// N = 8192 detections, 5 floats each. Output: 8192 x 5 floats (sorted order).
// ---------------------------------------------------------------------------

#define NDET 8192
#define NW   256          // 32-bit words per mask row (8192/32)
#define IMGS 640.0f

typedef float v2f __attribute__((ext_vector_type(2)));
typedef float v8f __attribute__((ext_vector_type(8)));

__device__ __forceinline__ unsigned ballot32(bool p) {
  return __builtin_amdgcn_ballot_w32(p);
}

// ---------------------------------------------------------------------------
// Kernel 1: max over the 8192 confidences (raw[j*5+4], all positive).
// ---------------------------------------------------------------------------
__global__ __launch_bounds__(256) void rtdetr_maxconf(const float* __restrict__ raw,
                                                      float* __restrict__ maxc) {
  __shared__ float sm[256];
  const int t = threadIdx.x;
  float m = 0.0f;
  #pragma unroll
  for (int e = 0; e < 32; ++e) {
    const int idx = t + e * 256;
    m = fmaxf(m, raw[idx * 5 + 4]);
  }
  sm[t] = m;
  __syncthreads();
  for (int s = 128; s > 0; s >>= 1) {
    if (t < s) sm[t] = fmaxf(sm[t], sm[t + s]);
    __syncthreads();
  }
  if (t == 0) *maxc = sm[0];
}

// ---------------------------------------------------------------------------
// Kernel 2: single-workgroup LDS bitonic sort (descending by conf) of packed
// 64-bit keys (conf_bits<<32 | idx), then gather/convert boxes into SoA:
// x1,y1,x2,y2, area, score (conf / maxconf).  64 KB static LDS.
// ---------------------------------------------------------------------------
__global__ __launch_bounds__(1024) void rtdetr_sort_prep(
    const float* __restrict__ raw, const float* __restrict__ maxc,
    float* __restrict__ bx1, float* __restrict__ by1,
    float* __restrict__ bx2, float* __restrict__ by2,
    float* __restrict__ area, float* __restrict__ score) {
  __shared__ unsigned long long key[NDET];   // 64 KB
  const int t = threadIdx.x;

  #pragma unroll
  for (int e = 0; e < 8; ++e) {
    const int s = t + e * 1024;
    const unsigned fb = __float_as_uint(raw[s * 5 + 4]);   // positive floats: bits are order-preserving
    key[s] = ((unsigned long long)fb << 32) | (unsigned)s;
  }
  __syncthreads();

  for (int k = 2; k <= NDET; k <<= 1) {
    for (int j = k >> 1; j > 0; j >>= 1) {
      #pragma unroll
      for (int e = 0; e < 8; ++e) {
        const int i = t + e * 1024;
        const int p = i ^ j;
        if (p > i) {
          const bool desc = ((i & k) == 0);   // final merge (k==NDET) -> fully descending
          const unsigned long long a = key[i];
          const unsigned long long b = key[p];
          const bool doswap = desc ? (a < b) : (a > b);
          if (doswap) { key[i] = b; key[p] = a; }
        }
      }
      __syncthreads();
    }
  }

  const float mc = *maxc;
  #pragma unroll
  for (int e = 0; e < 8; ++e) {
    const int s = t + e * 1024;
    const unsigned long long kv = key[s];
    const int idx = (int)(kv & 0xFFFFFFFFull);
    const float conf = __uint_as_float((unsigned)(kv >> 32));
    const float cx = raw[idx * 5 + 0] * IMGS;
    const float cy = raw[idx * 5 + 1] * IMGS;
    const float w  = raw[idx * 5 + 2] * IMGS;
    const float h  = raw[idx * 5 + 3] * IMGS;
    const float x1 = cx - 0.5f * w, y1 = cy - 0.5f * h;
    const float x2 = cx + 0.5f * w, y2 = cy + 0.5f * h;
    bx1[s] = x1; by1[s] = y1; bx2[s] = x2; by2[s] = y2;
    area[s]  = fmaxf(x2 - x1, 0.0f) * fmaxf(y2 - y1, 0.0f);
    score[s] = conf / mc;
  }
}

// ---------------------------------------------------------------------------
// Kernel 3: suppression bitmask.  One wave32 per 16-row x 32-col block.
// D[m][n] = area_i[m] + area_j[n] via V_WMMA_F32_16X16X4_F32 (rank-2 outer
// sum:  A rows = [area_i, 1, 0, 0],  B rows = [1...; area_j...; 0; 0]).
// IoU>0.5  <=>  3*inter > area_i+area_j   (division-free).
// Ballot packs 16 cols x 2 row-halves per D register into mask words.
// ---------------------------------------------------------------------------
__global__ __launch_bounds__(32) void rtdetr_supmask(
    const float* __restrict__ bx1, const float* __restrict__ by1,
    const float* __restrict__ bx2, const float* __restrict__ by2,
    const float* __restrict__ area, unsigned* __restrict__ mask) {
  const int lane = threadIdx.x;            // 0..31
  const int r0 = blockIdx.y * 16;          // first row of block
  const int c0 = blockIdx.x * 32;          // first col of block (one mask word)
  const int wb = blockIdx.x;               // word index within row

  if (c0 + 31 <= r0) {                     // entire block has j <= i: mask = 0
    if (lane < 16) mask[(size_t)(r0 + lane) * NW + wb] = 0u;
    return;
  }

  const int half = lane >> 4;              // 0 -> lanes 0-15, 1 -> lanes 16-31
  const int ln   = lane & 15;

  // A (16x4 f32): lanes 0-15 hold M=0..15 with {K0,K1}; lanes 16-31 hold {K2,K3}=0
  v2f av;
  av.x = half ? 0.0f : area[r0 + ln];
  av.y = half ? 0.0f : 1.0f;
  // B (4x16 f32): lanes 0-15 hold N=0..15 with {K0,K1}; lanes 16-31 hold {K2,K3}=0
  v2f b0v, b1v;
  b0v.x = half ? 0.0f : 1.0f;
  b0v.y = half ? 0.0f : area[c0 + ln];
  b1v.x = b0v.x;
  b1v.y = half ? 0.0f : area[c0 + 16 + ln];

  v8f cz = {};
  // areasum tiles: d0 -> cols c0..c0+15, d1 -> cols c0+16..c0+31
  v8f d0 = __builtin_amdgcn_wmma_f32_16x16x4_f32(false, av, false, b0v, (short)0, cz, false, false);
  v8f d1 = __builtin_amdgcn_wmma_f32_16x16x4_f32(false, av, false, b1v, (short)0, cz, false, false);

  // This lane's column boxes (one per tile)
  const int n0 = c0 + ln, n1 = c0 + 16 + ln;
  const float c0x1 = bx1[n0], c0y1 = by1[n0], c0x2 = bx2[n0], c0y2 = by2[n0];
  const float c1x1 = bx1[n1], c1y1 = by1[n1], c1x2 = bx2[n1], c1y2 = by2[n1];

  unsigned bal0[8], bal1[8];
  #pragma unroll
  for (int r = 0; r < 8; ++r) {
    const int mrow = r0 + half * 8 + r;    // D VGPR r: rows r (lanes 0-15) / r+8 (lanes 16-31)
    const float rx1 = bx1[mrow], ry1 = by1[mrow];
    const float rx2 = bx2[mrow], ry2 = by2[mrow];

    float iw0 = fminf(rx2, c0x2) - fmaxf(rx1, c0x1);
    float ih0 = fminf(ry2, c0y2) - fmaxf(ry1, c0y1);
    const float inter0 = fmaxf(iw0, 0.0f) * fmaxf(ih0, 0.0f);
    const bool p0 = (3.0f * inter0 > d0[r]) && (n0 > mrow);

    float iw1 = fminf(rx2, c1x2) - fmaxf(rx1, c1x1);
    float ih1 = fminf(ry2, c1y2) - fmaxf(ry1, c1y1);
    const float inter1 = fmaxf(iw1, 0.0f) * fmaxf(ih1, 0.0f);
    const bool p1 = (3.0f * inter1 > d1[r]) && (n1 > mrow);

    bal0[r] = ballot32(p0);
    bal1[r] = ballot32(p1);
  }

  if (lane == 0) {
    #pragma unroll
    for (int r = 0; r < 8; ++r) {
      // low 16 bits of ballot = rows r (cols c0..+15); high 16 = rows r+8
      const unsigned lo = (bal0[r] & 0xFFFFu) | (bal1[r] << 16);
      const unsigned hi = (bal0[r] >> 16) | (bal1[r] & 0xFFFF0000u);
      mask[(size_t)(r0 + r) * NW + wb]     = lo;
      mask[(size_t)(r0 + 8 + r) * NW + wb] = hi;
    }
  }
}

// ---------------------------------------------------------------------------
// Kernel 4: sequential greedy NMS scan — one wave32, remv[256] in LDS.
// For each surviving i (in sorted order), OR its mask row into remv.
// ---------------------------------------------------------------------------
__global__ __launch_bounds__(32) void rtdetr_nms_reduce(const unsigned* __restrict__ mask,
                                                        unsigned* __restrict__ remw) {
  __shared__ unsigned remv[NW];
  const int lane = threadIdx.x;
  #pragma unroll
  for (int s = 0; s < 8; ++s) remv[lane + 32 * s] = 0u;
  __syncthreads();

  for (int i = 0; i < NDET; ++i) {
    __builtin_prefetch(mask + (size_t)(i + 1) * NW + lane * 8, 0, 1);
    const unsigned removed = (remv[i >> 5] >> (i & 31)) & 1u;   // uniform broadcast read
    if (!removed) {
      const unsigned* row = mask + (size_t)i * NW;
      #pragma unroll
      for (int s = 0; s < 8; ++s)
        remv[lane + 32 * s] |= row[lane + 32 * s];
    }
    __syncthreads();
  }

  #pragma unroll
  for (int s = 0; s < 8; ++s) remw[lane + 32 * s] = remv[lane + 32 * s];
}

// ---------------------------------------------------------------------------
// Kernel 5: final where(keep & score>=0.25, [boxes score], 0)
// ---------------------------------------------------------------------------
__global__ __launch_bounds__(256) void rtdetr_finalize(
    const float* __restrict__ bx1, const float* __restrict__ by1,
    const float* __restrict__ bx2, const float* __restrict__ by2,
    const float* __restrict__ score, const unsigned* __restrict__ remw,
    float* __restrict__ out) {
  const int s = blockIdx.x * blockDim.x + threadIdx.x;
  if (s >= NDET) return;
  const bool kept = (((remw[s >> 5] >> (s & 31)) & 1u) == 0u) && (score[s] >= 0.25f);
  out[s * 5 + 0] = kept ? bx1[s]   : 0.0f;
  out[s * 5 + 1] = kept ? by1[s]   : 0.0f;
  out[s * 5 + 2] = kept ? bx2[s]   : 0.0f;
  out[s * 5 + 3] = kept ? by2[s]   : 0.0f;
  out[s * 5 + 4] = kept ? score[s] : 0.0f;
}

// ---------------------------------------------------------------------------
extern "C" void kernel_launch(void* const* d_in, const int* in_sizes, int n_in,
                              void* d_out, int out_size, void* d_ws, size_t ws_size,
                              hipStream_t stream) {
  (void)in_sizes; (void)n_in; (void)out_size; (void)ws_size;
  const float* raw = (const float*)d_in[0];
  float* F = (float*)d_ws;
  float* bx1   = F;
  float* by1   = F + 1 * NDET;
  float* bx2   = F + 2 * NDET;
  float* by2   = F + 3 * NDET;
  float* area  = F + 4 * NDET;
  float* score = F + 5 * NDET;
  float* maxc  = F + 6 * NDET;                      // 1 float (+padding)
  unsigned* mask = (unsigned*)(F + 8 * NDET);       // 8192*256 words = 8 MB
  unsigned* remw = mask + (size_t)NDET * NW;        // 256 words
  float* out = (float*)d_out;

  rtdetr_maxconf<<<1, 256, 0, stream>>>(raw, maxc);
  rtdetr_sort_prep<<<1, 1024, 0, stream>>>(raw, maxc, bx1, by1, bx2, by2, area, score);
  dim3 grid(NDET / 32, NDET / 16);                  // 256 x 512 wave-blocks
  rtdetr_supmask<<<grid, 32, 0, stream>>>(bx1, by1, bx2, by2, area, mask);
  rtdetr_nms_reduce<<<1, 32, 0, stream>>>(mask, remw);
  rtdetr_finalize<<<NDET / 256, 256, 0, stream>>>(bx1, by1, bx2, by2, score, remw, out);
}